// LieOptPoseEstimator_59742995087661
// MI455X (gfx1250) — compile-verified
//
#include <hip/hip_runtime.h>
#include <math.h>

typedef __attribute__((ext_vector_type(2))) float v2f;
typedef __attribute__((ext_vector_type(4))) float v4f;
typedef __attribute__((ext_vector_type(8))) float v8f;

#define NWAVES 8
#define BLOCK 256
#define ROWSTRIDE 16        // floats per staged K-slice (two 8-wide row streams)
#define SLICES_PER_WAVE 48  // 96 rows per 32-point chunk, 2 rows packed per slice
#define N_ITERS 10
#define DAMP 1e-8f
#define SQRT2F 1.41421356237309515f

__launch_bounds__(BLOCK)
__global__ void lieopt_gn_kernel(const float* __restrict__ src4,
                                 const float* __restrict__ trg4,
                                 const float* __restrict__ wgt,
                                 const float* __restrict__ Tinit,
                                 const float* __restrict__ invcov,
                                 const float* __restrict__ Tsv,
                                 float* __restrict__ out,
                                 int N)
{
    // 24 KB staging + 8 KB reduction + 1 KB Gram + pose: ~33.3 KB LDS
    __shared__ float stage[NWAVES * SLICES_PER_WAVE * ROWSTRIDE];
    __shared__ float redBuf[NWAVES * 256];
    __shared__ float Dmat[256];
    __shared__ float Ts[16];

    const int tid  = threadIdx.x;
    const int lane = tid & 31;
    const int wave = tid >> 5;
    const int b    = blockIdx.x;

    if (tid < 16) Ts[tid] = Tinit[(size_t)b * 16 + tid];

    const float* srcB = src4   + (size_t)b * 4 * N;
    const float* trgB = trg4   + (size_t)b * 4 * N;
    const float* wB   = wgt    + (size_t)b * N;
    const float* McB  = invcov + (size_t)b * N * 9;

    float* stW = stage + wave * SLICES_PER_WAVE * ROWSTRIDE;
    const int nChunks = (N + BLOCK - 1) / BLOCK;  // 32-point chunks handled by this wave

    const int col   = lane & 15;          // Gram column this lane owns (A/B operand layout)
    const int rsel2 = (lane >> 4) << 1;   // lanes 16..31 hold K-slices 2,3

    // Row-stream packing: lanes 0..15 write stream A (slice cols 0..7),
    // lanes 16..31 write stream B (slice cols 8..15). Every slice column is
    // rewritten each chunk; cols 7 and 15 are stored as explicit zeros so the
    // 7-wide [J|r] rows pad cleanly into each 8-wide half-tile.
    float* rp0 = stW + (3 * (lane & 15)) * ROWSTRIDE + ((lane >> 4) << 3);

#pragma unroll 1
    for (int it = 0; it < N_ITERS; ++it) {
        __syncthreads();  // pose from previous iteration (or init) visible to all
        const float R00=Ts[0], R01=Ts[1], R02=Ts[2],  t0=Ts[3];
        const float R10=Ts[4], R11=Ts[5], R12=Ts[6],  t1=Ts[7];
        const float R20=Ts[8], R21=Ts[9], R22=Ts[10], t2=Ts[11];

        v8f acc0 = {0.f,0.f,0.f,0.f,0.f,0.f,0.f,0.f};
        v8f acc1 = {0.f,0.f,0.f,0.f,0.f,0.f,0.f,0.f};

        for (int c = 0; c < nChunks; ++c) {
            const int n   = (c * NWAVES + wave) * 32 + lane;
            const bool ok = (n < N);
            const int nc  = ok ? n : 0;
            const float msk = ok ? 1.0f : 0.0f;

            const float s0 = srcB[0*N+nc], s1 = srcB[1*N+nc], s2 = srcB[2*N+nc];
            const float g0 = trgB[0*N+nc], g1 = trgB[1*N+nc], g2 = trgB[2*N+nc];
            const float wn = wB[nc] * msk;  // wn==0 zeroes the whole row triple
            const float* Mp = McB + (size_t)nc * 9;
            const float m00=Mp[0], m10=Mp[3], m11=Mp[4], m20=Mp[6], m21=Mp[7], m22=Mp[8];

            // C = sqrt(2)*w * chol(inv_cov)^T  (upper triangular whitener; C^T C = 2 w^2 inv_cov)
            const float l00 = sqrtf(m00);
            const float i00 = 1.0f / l00;
            const float l10 = m10 * i00, l20 = m20 * i00;
            const float l11 = sqrtf(m11 - l10 * l10);
            const float i11 = 1.0f / l11;
            const float l21 = (m21 - l20 * l10) * i11;
            const float l22 = sqrtf(fmaxf(m22 - l20 * l20 - l21 * l21, 0.0f));
            const float kk = SQRT2F * wn;
            const float c00=kk*l00, c01=kk*l10, c02=kk*l20, c11=kk*l11, c12=kk*l21, c22=kk*l22;

            // P = C*R (triangular x full), Q = P*hat(s); J = [-P | Q], r = C*(trg - R s - t)
            const float P00 = c00*R00 + c01*R10 + c02*R20;
            const float P01 = c00*R01 + c01*R11 + c02*R21;
            const float P02 = c00*R02 + c01*R12 + c02*R22;
            const float P10 = c11*R10 + c12*R20;
            const float P11 = c11*R11 + c12*R21;
            const float P12 = c11*R12 + c12*R22;
            const float P20 = c22*R20, P21 = c22*R21, P22 = c22*R22;

            const float Q00 = P01*s2 - P02*s1, Q01 = P02*s0 - P00*s2, Q02 = P00*s1 - P01*s0;
            const float Q10 = P11*s2 - P12*s1, Q11 = P12*s0 - P10*s2, Q12 = P10*s1 - P11*s0;
            const float Q20 = P21*s2 - P22*s1, Q21 = P22*s0 - P20*s2, Q22 = P20*s1 - P21*s0;

            const float e0 = g0 - (R00*s0 + R01*s1 + R02*s2 + t0);
            const float e1 = g1 - (R10*s0 + R11*s1 + R12*s2 + t1);
            const float e2 = g2 - (R20*s0 + R21*s1 + R22*s2 + t2);
            const float r0 = c00*e0 + c01*e1 + c02*e2;
            const float r1 = c11*e1 + c12*e2;
            const float r2 = c22*e2;

            // Stage this point's three 7-wide rows [J | r] into this lane's stream half
            *(v4f*)(rp0 +  0) = (v4f){-P00, -P01, -P02, Q00};
            *(v4f*)(rp0 +  4) = (v4f){ Q01,  Q02,  r0,  0.f};
            *(v4f*)(rp0 + 16) = (v4f){-P10, -P11, -P12, Q10};
            *(v4f*)(rp0 + 20) = (v4f){ Q11,  Q12,  r1,  0.f};
            *(v4f*)(rp0 + 32) = (v4f){-P20, -P21, -P22, Q20};
            *(v4f*)(rp0 + 36) = (v4f){ Q21,  Q22,  r2,  0.f};
            // Same-wave DS ordering: writes complete before the reads below (dscnt).

            // Gram accumulation, K=4 slices per WMMA, 2 packed row streams per slice.
            // A layout (f32 16x4): lane = M (=Gram column), VGPR0 = K 0/2, VGPR1 = K 1/3.
            // Symmetric Gram => B per-lane data equals A's; pass `a` twice.
            // Two accumulators break the C->D serial chain.
#pragma unroll
            for (int j = 0; j < 12; j += 2) {
                const float* ap0 = stW + (4 * j     + rsel2) * ROWSTRIDE + col;
                const float* ap1 = stW + (4 * (j+1) + rsel2) * ROWSTRIDE + col;
                v2f a0, a1;
                a0.x = ap0[0]; a0.y = ap0[ROWSTRIDE];
                a1.x = ap1[0]; a1.y = ap1[ROWSTRIDE];
                acc0 = __builtin_amdgcn_wmma_f32_16x16x4_f32(
                         false, a0, false, a0, (short)0, acc0, false, false);
                acc1 = __builtin_amdgcn_wmma_f32_16x16x4_f32(
                         false, a1, false, a1, (short)0, acc1, false, false);
            }
        }

        const v8f acc = acc0 + acc1;

        // Reduce the 8 per-wave 16x16 accumulators into Dmat[m][n]
#pragma unroll
        for (int v = 0; v < 8; ++v) redBuf[wave * 256 + v * 32 + lane] = acc[v];
        __syncthreads();
        {
            const int v = tid >> 5, l = tid & 31;
            float sum = 0.0f;
            for (int ww = 0; ww < NWAVES; ++ww) sum += redBuf[ww * 256 + v * 32 + l];
            const int mrow = v + ((l >> 4) << 3);  // C/D layout: VGPR v -> M=v / v+8
            const int ncol = l & 15;
            Dmat[mrow * 16 + ncol] = sum;
        }
        __syncthreads();

        if (tid == 0) {
            // Combine the two stream diagonal blocks:
            // H = D[0:6,0:6] + D[8:14,8:14] + damp*I ; g = D[0:6,6] + D[8:14,14]
            float Am[6][7];
            for (int i = 0; i < 6; ++i) {
                for (int j = 0; j < 6; ++j)
                    Am[i][j] = Dmat[i * 16 + j] + Dmat[(i + 8) * 16 + (j + 8)]
                             + (i == j ? DAMP : 0.0f);
                Am[i][6] = -(Dmat[i * 16 + 6] + Dmat[(i + 8) * 16 + 14]);
            }
            // Solve H dx = -g (SPD -> no pivoting)
            for (int k2 = 0; k2 < 6; ++k2) {
                const float p = 1.0f / Am[k2][k2];
                for (int j = k2; j < 7; ++j) Am[k2][j] *= p;
                for (int i = k2 + 1; i < 6; ++i) {
                    const float f = Am[i][k2];
                    for (int j = k2; j < 7; ++j) Am[i][j] -= f * Am[k2][j];
                }
            }
            float dx[6];
            for (int i = 5; i >= 0; --i) {
                float s = Am[i][6];
                for (int j = i + 1; j < 6; ++j) s -= Am[i][j] * dx[j];
                dx[i] = s;
            }

            // E = se3_exp(dx)
            const float px = dx[3], py = dx[4], pz = dx[5];
            const float th2 = px*px + py*py + pz*pz;
            float Aa, Bb, Cc;
            if (th2 < 1e-12f) { Aa = 1.0f; Bb = 0.5f; Cc = 1.0f / 6.0f; }
            else {
                const float th = sqrtf(th2), sn = sinf(th), cs = cosf(th);
                Aa = sn / th; Bb = (1.0f - cs) / th2; Cc = (th - sn) / (th2 * th);
            }
            const float KS00 = -(py*py + pz*pz), KS11 = -(px*px + pz*pz), KS22 = -(px*px + py*py);
            const float KS01 = px*py, KS02 = px*pz, KS12 = py*pz;
            const float E00 = 1.f + Bb*KS00,     E01 = -Aa*pz + Bb*KS01, E02 =  Aa*py + Bb*KS02;
            const float E10 =  Aa*pz + Bb*KS01,  E11 = 1.f + Bb*KS11,    E12 = -Aa*px + Bb*KS12;
            const float E20 = -Aa*py + Bb*KS02,  E21 =  Aa*px + Bb*KS12, E22 = 1.f + Bb*KS22;
            const float V00 = 1.f + Cc*KS00,     V01 = -Bb*pz + Cc*KS01, V02 =  Bb*py + Cc*KS02;
            const float V10 =  Bb*pz + Cc*KS01,  V11 = 1.f + Cc*KS11,    V12 = -Bb*px + Cc*KS12;
            const float V20 = -Bb*py + Cc*KS02,  V21 =  Bb*px + Cc*KS12, V22 = 1.f + Cc*KS22;
            const float te0 = V00*dx[0] + V01*dx[1] + V02*dx[2];
            const float te1 = V10*dx[0] + V11*dx[1] + V12*dx[2];
            const float te2 = V20*dx[0] + V21*dx[1] + V22*dx[2];

            // T = T @ E (rigid compose)
            const float a00=Ts[0],a01=Ts[1],a02=Ts[2], b0=Ts[3];
            const float a10=Ts[4],a11=Ts[5],a12=Ts[6], b1=Ts[7];
            const float a20=Ts[8],a21=Ts[9],a22=Ts[10],b2=Ts[11];
            Ts[0]=a00*E00+a01*E10+a02*E20; Ts[1]=a00*E01+a01*E11+a02*E21; Ts[2]=a00*E02+a01*E12+a02*E22;
            Ts[3]=a00*te0+a01*te1+a02*te2+b0;
            Ts[4]=a10*E00+a11*E10+a12*E20; Ts[5]=a10*E01+a11*E11+a12*E21; Ts[6]=a10*E02+a11*E12+a12*E22;
            Ts[7]=a10*te0+a11*te1+a12*te2+b1;
            Ts[8]=a20*E00+a21*E10+a22*E20; Ts[9]=a20*E01+a21*E11+a22*E21; Ts[10]=a20*E02+a21*E12+a22*E22;
            Ts[11]=a20*te0+a21*te1+a22*te2+b2;
            Ts[12]=0.f; Ts[13]=0.f; Ts[14]=0.f; Ts[15]=1.f;
        }
    }

    __syncthreads();
    if (tid == 0) {
        // out = inv(Tsv) @ T @ Tsv  (all rigid)
        const float S00=Tsv[0], S01=Tsv[1], S02=Tsv[2],  sx=Tsv[3];
        const float S10=Tsv[4], S11=Tsv[5], S12=Tsv[6],  sy=Tsv[7];
        const float S20=Tsv[8], S21=Tsv[9], S22=Tsv[10], sz=Tsv[11];
        const float R00=Ts[0],R01=Ts[1],R02=Ts[2], t0=Ts[3];
        const float R10=Ts[4],R11=Ts[5],R12=Ts[6], t1=Ts[7];
        const float R20=Ts[8],R21=Ts[9],R22=Ts[10],t2=Ts[11];
        // M = T @ Tsv
        const float M00=R00*S00+R01*S10+R02*S20, M01=R00*S01+R01*S11+R02*S21, M02=R00*S02+R01*S12+R02*S22;
        const float M10=R10*S00+R11*S10+R12*S20, M11=R10*S01+R11*S11+R12*S21, M12=R10*S02+R11*S12+R12*S22;
        const float M20=R20*S00+R21*S10+R22*S20, M21=R20*S01+R21*S11+R22*S21, M22=R20*S02+R21*S12+R22*S22;
        const float Mt0=R00*sx+R01*sy+R02*sz+t0;
        const float Mt1=R10*sx+R11*sy+R12*sz+t1;
        const float Mt2=R20*sx+R21*sy+R22*sz+t2;
        // O = Tsv^-1 @ M : Orot = S^T Mrot ; Ot = S^T (Mt - s)
        const float d0=Mt0-sx, d1=Mt1-sy, d2=Mt2-sz;
        float* ob = out + (size_t)b * 16;
        ob[0]=S00*M00+S10*M10+S20*M20; ob[1]=S00*M01+S10*M11+S20*M21; ob[2]=S00*M02+S10*M12+S20*M22;
        ob[3]=S00*d0+S10*d1+S20*d2;
        ob[4]=S01*M00+S11*M10+S21*M20; ob[5]=S01*M01+S11*M11+S21*M21; ob[6]=S01*M02+S11*M12+S21*M22;
        ob[7]=S01*d0+S11*d1+S21*d2;
        ob[8]=S02*M00+S12*M10+S22*M20; ob[9]=S02*M01+S12*M11+S22*M21; ob[10]=S02*M02+S12*M12+S22*M22;
        ob[11]=S02*d0+S12*d1+S22*d2;
        ob[12]=0.f; ob[13]=0.f; ob[14]=0.f; ob[15]=1.f;
    }
}

extern "C" void kernel_launch(void* const* d_in, const int* in_sizes, int n_in,
                              void* d_out, int out_size, void* d_ws, size_t ws_size,
                              hipStream_t stream) {
    const float* src4   = (const float*)d_in[0];  // (B,4,N)
    const float* trg4   = (const float*)d_in[1];  // (B,4,N)
    const float* wgt    = (const float*)d_in[2];  // (B,1,N)
    const float* Tinit  = (const float*)d_in[3];  // (B,4,4)
    const float* invcov = (const float*)d_in[4];  // (B,N,3,3)
    const float* Tsv    = (const float*)d_in[5];  // (4,4)

    const int B = in_sizes[3] / 16;
    const int N = in_sizes[2] / B;

    lieopt_gn_kernel<<<B, BLOCK, 0, stream>>>(src4, trg4, wgt, Tinit, invcov, Tsv,
                                              (float*)d_out, N);
}